// VectorQuantizer_35974646071763
// MI455X (gfx1250) — compile-verified
//
#include <hip/hip_runtime.h>

typedef float v2f __attribute__((ext_vector_type(2)));
typedef float v4f __attribute__((ext_vector_type(4)));
typedef float v8f __attribute__((ext_vector_type(8)));

#define NS 16384          // flattened samples 32*512
#define NE 2048           // codebook size
#define ED 128            // embedding dim
#define SKI 100           // sinkhorn iterations
#define INV_EPS 100.0f    // 1/0.01
#define ONE_PLUS_BETA 1.25f

// ---- d_out layout (floats): x_q | loss | indices | d ----
#define OFF_LOSS ((size_t)NS * ED)
#define OFF_IDX  (OFF_LOSS + 1)
#define OFF_D    (OFF_IDX + NS)

// ---- workspace layout (floats): Q0 first (aligned), then small vectors ----
#define WS_XN   ((size_t)NE * NS)
#define WS_EN   (WS_XN + NS)
#define WS_MM   (WS_EN + NE)
#define WS_CMIN (WS_MM + 8192)
#define WS_MK   (WS_CMIN + NE)
#define WS_SC   (WS_MK + NE)
#define WS_U    (WS_SC + 4)
#define WS_V    (WS_U + NE)
#define WS_IDX  (WS_V + NS)
#define WS_LP   (WS_IDX + NS)

__device__ __forceinline__ unsigned fkey(float f) {
  unsigned u = __float_as_uint(f);
  return (u & 0x80000000u) ? ~u : (u | 0x80000000u);   // order-preserving uint key
}
__device__ __forceinline__ float funkey(unsigned k) {
  unsigned u = (k & 0x80000000u) ? (k & 0x7fffffffu) : ~k;
  return __uint_as_float(u);
}

// ---- row norms of x (16384 rows) and emb (2048 rows); one wave per row ----
__global__ __launch_bounds__(256) void k_norms(const float* __restrict__ X,
                                               const float* __restrict__ E,
                                               float* __restrict__ xn,
                                               float* __restrict__ en) {
  const int t = threadIdx.x, wave = t >> 5, lane = t & 31;
  const int row = blockIdx.x * 8 + wave;
  const float* src;
  float* dst;
  if (row < NS) { src = X + (size_t)row * ED; dst = xn + row; }
  else          { src = E + (size_t)(row - NS) * ED; dst = en + (row - NS); }
  v4f a = *(const v4f*)(src + lane * 4);
  float s = a.x * a.x + a.y * a.y + a.z * a.z + a.w * a.w;
  for (int o = 16; o > 0; o >>= 1) s += __shfl_xor(s, o, 32);
  if (lane == 0) *dst = s;
}

__global__ __launch_bounds__(256) void k_init(unsigned* __restrict__ cmin) {
  cmin[blockIdx.x * 256 + threadIdx.x] = 0xFFFFFFFFu;
}

// ---- WMMA GEMM: d[m][n] = |x_m|^2 + |e_n|^2 - 2 x_m.e_n ; raw d -> d_out section,
//      per-block min/max partials -> mm. Each wave: 16(M)x64(N) tile, K=128 via 16x16x4 f32 WMMA.
__global__ __launch_bounds__(256) void k_gemm(const float* __restrict__ X,
                                              const float* __restrict__ E,
                                              const float* __restrict__ xn,
                                              const float* __restrict__ en,
                                              float* __restrict__ dR,
                                              float* __restrict__ mm) {
  __shared__ float smn[8], smx[8];
  const int t = threadIdx.x, wave = t >> 5, lane = t & 31;
  const int gw = blockIdx.x * 8 + wave;
  const int m0 = (gw >> 5) << 4;     // 1024 M-tiles
  const int n0 = (gw & 31) << 6;     // 32 N-groups of 64
  const int hi = lane >> 4;          // K sub-pair select per ISA layout
  const int l15 = lane & 15;
  const float* arow = X + (size_t)(m0 + l15) * ED;
  const float* b0r = E + (size_t)(n0 + l15) * ED;
  const float* b1r = b0r + 16 * ED;
  const float* b2r = b0r + 32 * ED;
  const float* b3r = b0r + 48 * ED;
  v8f acc[4] = {v8f{0.f}, v8f{0.f}, v8f{0.f}, v8f{0.f}};
  for (int k = 0; k < ED; k += 4) {
    const int ko = k + hi * 2;
    v2f a  = *(const v2f*)(arow + ko);
    v2f b0 = *(const v2f*)(b0r + ko);
    v2f b1 = *(const v2f*)(b1r + ko);
    v2f b2 = *(const v2f*)(b2r + ko);
    v2f b3 = *(const v2f*)(b3r + ko);
    acc[0] = __builtin_amdgcn_wmma_f32_16x16x4_f32(false, a, false, b0, (short)0, acc[0], false, false);
    acc[1] = __builtin_amdgcn_wmma_f32_16x16x4_f32(false, a, false, b1, (short)0, acc[1], false, false);
    acc[2] = __builtin_amdgcn_wmma_f32_16x16x4_f32(false, a, false, b2, (short)0, acc[2], false, false);
    acc[3] = __builtin_amdgcn_wmma_f32_16x16x4_f32(false, a, false, b3, (short)0, acc[3], false, false);
  }
  float vmn = 3.4e38f, vmx = -3.4e38f;
#pragma unroll
  for (int j = 0; j < 4; ++j) {
    const int col = n0 + j * 16 + l15;
    const float ec = en[col];
#pragma unroll
    for (int i = 0; i < 8; ++i) {
      const int row = m0 + hi * 8 + i;           // C layout: VGPR i -> M=i / M=i+8
      const float val = xn[row] + ec - 2.0f * acc[j][i];
      dR[(size_t)row * NE + col] = val;          // b32 store (d section only 4B aligned)
      vmn = fminf(vmn, val);
      vmx = fmaxf(vmx, val);
    }
  }
  for (int o = 16; o > 0; o >>= 1) {
    vmn = fminf(vmn, __shfl_xor(vmn, o, 32));
    vmx = fmaxf(vmx, __shfl_xor(vmx, o, 32));
  }
  if (lane == 0) { smn[wave] = vmn; smx[wave] = vmx; }
  __syncthreads();
  if (t == 0) {
    float a = smn[0], b = smx[0];
    for (int w = 1; w < 8; ++w) { a = fminf(a, smn[w]); b = fmaxf(b, smx[w]); }
    mm[blockIdx.x] = a;
    mm[4096 + blockIdx.x] = b;
  }
}

__global__ __launch_bounds__(256) void k_mmfinal(const float* __restrict__ mm,
                                                 float* __restrict__ sc) {
  __shared__ float smn[256], smx[256];
  const int t = threadIdx.x;
  float mn = 3.4e38f, mx = -3.4e38f;
  for (int i = t; i < 4096; i += 256) { mn = fminf(mn, mm[i]); mx = fmaxf(mx, mm[4096 + i]); }
  smn[t] = mn; smx[t] = mx; __syncthreads();
  for (int o = 128; o > 0; o >>= 1) {
    if (t < o) { smn[t] = fminf(smn[t], smn[t + o]); smx[t] = fmaxf(smx[t], smx[t + o]); }
    __syncthreads();
  }
  if (t == 0) {
    const float mid = (smx[0] + smn[0]) * 0.5f;
    const float amp = smx[0] - mid + 1e-5f;
    sc[0] = mid;
    sc[1] = 1.0f / amp;
  }
}

// ---- center d in place (if use_sk) + per-codebook-column min (keyed atomicMin) ----
__global__ __launch_bounds__(256) void k_center(float* __restrict__ dC,
                                                const float* __restrict__ sc,
                                                const int* __restrict__ usk,
                                                unsigned* __restrict__ cmin) {
  const int t = threadIdx.x;
  const int k = (blockIdx.x & 7) * 256 + t;
  const int nb = (blockIdx.x >> 3) * 256;
  const float mid = sc[0], invA = sc[1];
  const int u = *usk;
  float lmin = 3.4e38f;
  for (int j = 0; j < 256; ++j) {
    const size_t off = (size_t)(nb + j) * NE + k;
    const float dv = dC[off];
    const float c = u ? (dv - mid) * invA : dv;
    dC[off] = c;
    lmin = fminf(lmin, c);
  }
  atomicMin(&cmin[k], fkey(lmin));
}

__global__ __launch_bounds__(256) void k_mk(const unsigned* __restrict__ cmin,
                                            float* __restrict__ mk) {
  const int k = blockIdx.x * 256 + threadIdx.x;
  mk[k] = -funkey(cmin[k]) * INV_EPS;   // max_n(-dc/eps); exp arg <= 0
}

// ---- Q0[k][n] = exp(-dc[n][k]/eps - mk[k]) via 64x64 LDS transpose tile ----
__global__ __launch_bounds__(256) void k_transexp(const float* __restrict__ dC,
                                                  const float* __restrict__ mk,
                                                  float* __restrict__ Q0) {
  __shared__ float tile[64][65];
  const int t = threadIdx.x;
  const int nB = blockIdx.x * 64;
  const int kB = blockIdx.y * 64;
  const int c = t & 63, r0 = t >> 6;
#pragma unroll
  for (int p = 0; p < 16; ++p) {
    const int r = r0 + p * 4;
    tile[r][c] = dC[(size_t)(nB + r) * NE + kB + c];
  }
  __syncthreads();
#pragma unroll
  for (int p = 0; p < 16; ++p) {
    const int r = r0 + p * 4;                       // local k
    const float m = mk[kB + r];
    Q0[(size_t)(kB + r) * NS + nB + c] = expf(-tile[c][r] * INV_EPS - m);
  }
}

__global__ __launch_bounds__(256) void k_vinit(float* __restrict__ v) {
  v[blockIdx.x * 256 + threadIdx.x] = 1.0f;
}

// ---- u[k] = 1/(K * sum_n Q0[k][n] * v[n]) ; one block per row, float4 streams ----
__global__ __launch_bounds__(256) void k_u(const float* __restrict__ Q0,
                                           const float* __restrict__ v,
                                           float* __restrict__ u) {
  __shared__ float s[256];
  const int k = blockIdx.x, t = threadIdx.x;
  const v4f* q  = (const v4f*)(Q0 + (size_t)k * NS);
  const v4f* vv = (const v4f*)v;
  float acc = 0.f;
#pragma unroll
  for (int i = 0; i < 16; ++i) {
    const v4f a = q[t + 256 * i];
    const v4f b = vv[t + 256 * i];
    acc += a.x * b.x + a.y * b.y + a.z * b.z + a.w * b.w;
  }
  s[t] = acc; __syncthreads();
  for (int o = 128; o > 0; o >>= 1) { if (t < o) s[t] += s[t + o]; __syncthreads(); }
  if (t == 0) u[k] = 1.0f / ((float)NE * s[0]);
}

// ---- v[n] = 1/(B * sum_k Q0[k][n] * u[k]) ; 128 n per block, 8-way k split ----
__global__ __launch_bounds__(256) void k_v(const float* __restrict__ Q0,
                                           const float* __restrict__ u,
                                           float* __restrict__ v) {
  __shared__ float su[NE];
  __shared__ float sp[8][128];
  const int t = threadIdx.x;
  for (int i = t; i < NE; i += 256) su[i] = u[i];
  __syncthreads();
  const int lane = t & 31, ks = t >> 5;
  const int n4 = blockIdx.x * 128 + lane * 4;
  v4f acc = {0.f, 0.f, 0.f, 0.f};
  for (int k = ks; k < NE; k += 8) {
    const v4f q = *(const v4f*)(Q0 + (size_t)k * NS + n4);
    const float uk = su[k];
    acc.x += q.x * uk; acc.y += q.y * uk; acc.z += q.z * uk; acc.w += q.w * uk;
  }
  sp[ks][lane * 4 + 0] = acc.x; sp[ks][lane * 4 + 1] = acc.y;
  sp[ks][lane * 4 + 2] = acc.z; sp[ks][lane * 4 + 3] = acc.w;
  __syncthreads();
  if (t < 128) {
    float s = 0.f;
#pragma unroll
    for (int j = 0; j < 8; ++j) s += sp[j][t];
    v[blockIdx.x * 128 + t] = 1.0f / ((float)NS * s);
  }
}

// ---- indices: argmax_k u[k]*Q0[k][n] (sinkhorn) or argmin_k d[n][k] ----
__global__ __launch_bounds__(256) void k_argmax(const float* __restrict__ Q0,
                                                const float* __restrict__ u,
                                                const float* __restrict__ dC,
                                                const int* __restrict__ usk,
                                                int* __restrict__ idxOut) {
  __shared__ float su[NE];
  __shared__ float sv[8][128];
  __shared__ int si[8][128];
  const int t = threadIdx.x;
  if (*usk) {
    for (int i = t; i < NE; i += 256) su[i] = u[i];
    __syncthreads();
    const int lane = t & 31, ks = t >> 5;
    const int n4 = blockIdx.x * 128 + lane * 4;
    float bv[4] = {-3.4e38f, -3.4e38f, -3.4e38f, -3.4e38f};
    int bi[4] = {0, 0, 0, 0};
    for (int k = ks; k < NE; k += 8) {
      const v4f q = *(const v4f*)(Q0 + (size_t)k * NS + n4);
      const float uk = su[k];
      const float s0 = q.x * uk, s1 = q.y * uk, s2 = q.z * uk, s3 = q.w * uk;
      if (s0 > bv[0]) { bv[0] = s0; bi[0] = k; }
      if (s1 > bv[1]) { bv[1] = s1; bi[1] = k; }
      if (s2 > bv[2]) { bv[2] = s2; bi[2] = k; }
      if (s3 > bv[3]) { bv[3] = s3; bi[3] = k; }
    }
#pragma unroll
    for (int j = 0; j < 4; ++j) { sv[ks][lane * 4 + j] = bv[j]; si[ks][lane * 4 + j] = bi[j]; }
    __syncthreads();
    if (t < 128) {
      float best = -3.4e38f; int bidx = NE;
#pragma unroll
      for (int j = 0; j < 8; ++j) {
        const float vv = sv[j][t]; const int ii = si[j][t];
        if (vv > best || (vv == best && ii < bidx)) { best = vv; bidx = ii; }
      }
      idxOut[blockIdx.x * 128 + t] = bidx;
    }
  } else {
    if (t < 128) {
      const int n = blockIdx.x * 128 + t;
      const float* row = dC + (size_t)n * NE;
      float best = row[0]; int bidx = 0;
      for (int k = 1; k < NE; ++k) {
        const float dv = row[k];
        if (dv < best) { best = dv; bidx = k; }
      }
      idxOut[n] = bidx;
    }
  }
}

// ---- x_q gather + per-sample loss partial + index-as-float output ----
__global__ __launch_bounds__(128) void k_gather(const float* __restrict__ X,
                                                const float* __restrict__ E,
                                                const int* __restrict__ idx,
                                                float* __restrict__ out,
                                                float* __restrict__ lp) {
  __shared__ float s[128];
  const int n = blockIdx.x, t = threadIdx.x;
  const int i = idx[n];
  const float ev = E[(size_t)i * ED + t];
  const float xv = X[(size_t)n * ED + t];
  out[(size_t)n * ED + t] = ev;                  // straight-through == x_q numerically
  const float df = ev - xv;
  s[t] = df * df; __syncthreads();
  for (int o = 64; o > 0; o >>= 1) { if (t < o) s[t] += s[t + o]; __syncthreads(); }
  if (t == 0) { lp[n] = s[0]; out[OFF_IDX + n] = (float)i; }
}

__global__ __launch_bounds__(256) void k_lossfinal(const float* __restrict__ lp,
                                                   float* __restrict__ out) {
  __shared__ float s[256];
  const int t = threadIdx.x;
  float a = 0.f;
  for (int i = t; i < NS; i += 256) a += lp[i];
  s[t] = a; __syncthreads();
  for (int o = 128; o > 0; o >>= 1) { if (t < o) s[t] += s[t + o]; __syncthreads(); }
  if (t == 0) out[OFF_LOSS] = s[0] * (ONE_PLUS_BETA / (float)((size_t)NS * ED));
}

extern "C" void kernel_launch(void* const* d_in, const int* in_sizes, int n_in,
                              void* d_out, int out_size, void* d_ws, size_t ws_size,
                              hipStream_t stream) {
  const float* X = (const float*)d_in[0];
  const float* E = (const float*)d_in[1];
  const int* usk = (const int*)d_in[2];
  float* out = (float*)d_out;
  float* ws = (float*)d_ws;

  float* Q0 = ws;
  float* xn = ws + WS_XN;
  float* en = ws + WS_EN;
  float* mm = ws + WS_MM;
  unsigned* cmin = (unsigned*)(ws + WS_CMIN);
  float* mk = ws + WS_MK;
  float* sc = ws + WS_SC;
  float* u = ws + WS_U;
  float* v = ws + WS_V;
  int* idx = (int*)(ws + WS_IDX);
  float* lp = ws + WS_LP;
  float* dC = out + OFF_D;

  k_norms<<<(NS + NE) / 8, 256, 0, stream>>>(X, E, xn, en);
  k_init<<<NE / 256, 256, 0, stream>>>(cmin);
  k_gemm<<<4096, 256, 0, stream>>>(X, E, xn, en, dC, mm);
  k_mmfinal<<<1, 256, 0, stream>>>(mm, sc);
  k_center<<<512, 256, 0, stream>>>(dC, sc, usk, cmin);
  k_mk<<<NE / 256, 256, 0, stream>>>(cmin, mk);
  k_transexp<<<dim3(NS / 64, NE / 64), 256, 0, stream>>>(dC, mk, Q0);
  k_vinit<<<NS / 256, 256, 0, stream>>>(v);
  for (int it = 0; it < SKI; ++it) {
    k_u<<<NE, 256, 0, stream>>>(Q0, v, u);
    k_v<<<NS / 128, 256, 0, stream>>>(Q0, u, v);
  }
  k_argmax<<<NS / 128, 256, 0, stream>>>(Q0, u, dC, usk, idx);
  k_gather<<<NS, 128, 0, stream>>>(X, E, idx, out, lp);
  k_lossfinal<<<1, 256, 0, stream>>>(lp, out);
}